// Encoder_45938970198799
// MI455X (gfx1250) — compile-verified
//
#include <hip/hip_runtime.h>

typedef __attribute__((ext_vector_type(2))) float v2f;
typedef __attribute__((ext_vector_type(4))) float v4f;
typedef __attribute__((ext_vector_type(8))) float v8f;

#define NNODES 100000

// ---------------------------------------------------------------------------
// Degree / normalization
// ---------------------------------------------------------------------------
__global__ void init_deg_kernel(float* __restrict__ deg, int n) {
    int i = blockIdx.x * blockDim.x + threadIdx.x;
    if (i < n) deg[i] = 1.0f;  // self-loop contribution
}

__global__ void accum_deg_kernel(const int* __restrict__ dst, float* __restrict__ deg, int E) {
    int e = blockIdx.x * blockDim.x + threadIdx.x;
    if (e < E) atomicAdd(&deg[dst[e]], 1.0f);
}

__global__ void deg_to_dinv_kernel(float* __restrict__ deg, int n) {
    int i = blockIdx.x * blockDim.x + threadIdx.x;
    if (i < n) deg[i] = rsqrtf(deg[i]);  // deg >= 1 always (self-loops)
}

// ---------------------------------------------------------------------------
// Dense GEMM: C[M,N] = A[M,K] @ B[K,N] via V_WMMA_F32_16X16X4_F32.
// Register-blocked: one wave computes MB M-tiles x 1 N-tile (MB*16 x 16 out).
// Per K-step of 4: MB A-loads (b64) + 2 B-loads (b32) feed MB WMMAs, so the
// B operand is amortized MB-fold across the M dimension.
// A 16x4 lane layout (ISA 7.12.2): lanes 0-15 -> K={k,k+1}, lanes 16-31 ->
// K={k+2,k+3}: each lane loads a contiguous float2. B mirrored.
// ---------------------------------------------------------------------------
template <int MB>
__launch_bounds__(128)
__global__ void gemm_f32_wmma_kernel(const float* __restrict__ A,
                                     const float* __restrict__ B,
                                     float* __restrict__ C,
                                     int M, int N, int K) {
    const int lane = threadIdx.x & 31;
    const int wave = threadIdx.x >> 5;
    const int ntN  = N >> 4;              // N-tiles
    const int nMB  = M / (16 * MB);       // M-blocks
    const int job  = blockIdx.x * (blockDim.x >> 5) + wave;
    if (job >= nMB * ntN) return;         // wave-uniform: EXEC stays all-1s

    const int mb = job / ntN;
    const int nt = job - mb * ntN;
    const int m0 = mb * 16 * MB;
    const int n0 = nt << 4;

    const int l15   = lane & 15;
    const int khalf = (lane >> 4) << 1;   // 0 for lanes 0-15, 2 for lanes 16-31

    const float* abase = A + (size_t)(m0 + l15) * K + khalf;   // contiguous pair per lane
    const float* bcol  = B + (size_t)khalf * N + (n0 + l15);   // stride-N pair

    v8f acc[MB];
    #pragma unroll
    for (int t = 0; t < MB; ++t) acc[t] = (v8f){};

    for (int k = 0; k < K; k += 4) {
        v2f b;
        b.x = bcol[(size_t)k * N];
        b.y = bcol[(size_t)(k + 1) * N];
        #pragma unroll
        for (int t = 0; t < MB; ++t) {
            v2f a = *(const v2f*)(abase + (size_t)t * 16 * K + k);
            acc[t] = __builtin_amdgcn_wmma_f32_16x16x4_f32(
                /*neg_a=*/false, a, /*neg_b=*/false, b,
                /*c_mod=*/(short)0, acc[t], /*reuse_a=*/false, /*reuse_b=*/false);
        }
    }

    // C/D layout: VGPR r -> row +r (lanes 0-15) / +8+r (lanes 16-31), col n0+lane%16
    const int cmoff = (lane >> 4) << 3;
    #pragma unroll
    for (int t = 0; t < MB; ++t) {
        float* crow = C + (size_t)(m0 + t * 16 + cmoff) * N + (n0 + l15);
        #pragma unroll
        for (int r = 0; r < 8; ++r) crow[(size_t)r * N] = acc[t][r];
    }
}

// ---------------------------------------------------------------------------
// Aggregation init: out[i,f] = bias[f] + h[i,f] * dinv[i]^2   (self-loop term)
// ---------------------------------------------------------------------------
template <int F>
__global__ void agg_init_kernel(const float* __restrict__ h,
                                const float* __restrict__ dinv,
                                const float* __restrict__ bias,
                                float* __restrict__ out, int n) {
    int idx = blockIdx.x * blockDim.x + threadIdx.x;
    if (idx >= n * F) return;
    int i = idx >> (F == 128 ? 7 : 6);
    int f = idx & (F - 1);
    float di = dinv[i];
    out[idx] = bias[f] + h[idx] * di * di;
}

// ---------------------------------------------------------------------------
// Edge aggregation: one wave per edge. Vector gather from h[src] (stays in L2,
// 51.2 MB << 192 MB), scaled f32 atomics into out[dst] (L2 atomic units).
// ---------------------------------------------------------------------------
template <int F>
__launch_bounds__(256)
__global__ void agg_edges_kernel(const int* __restrict__ src,
                                 const int* __restrict__ dst,
                                 const float* __restrict__ dinv,
                                 const float* __restrict__ h,
                                 float* __restrict__ out, int E) {
    const int lane = threadIdx.x & 31;
    const int e    = (blockIdx.x * blockDim.x + threadIdx.x) >> 5;
    if (e >= E) return;

    const int   s    = src[e];
    const int   d    = dst[e];
    const float norm = dinv[s] * dinv[d];

    constexpr int VE = F / 32;  // floats per lane: 4 (F=128) or 2 (F=64)
    const float* hs = h   + (size_t)s * F + lane * VE;
    float*       od = out + (size_t)d * F + lane * VE;

    if constexpr (VE == 4) {
        v4f v = *(const v4f*)hs;  // global_load_b128
        atomicAdd(od + 0, v.x * norm);
        atomicAdd(od + 1, v.y * norm);
        atomicAdd(od + 2, v.z * norm);
        atomicAdd(od + 3, v.w * norm);
    } else {
        v2f v = *(const v2f*)hs;  // global_load_b64
        atomicAdd(od + 0, v.x * norm);
        atomicAdd(od + 1, v.y * norm);
    }
}

__global__ void relu_kernel(float* __restrict__ x, int n) {
    int i = blockIdx.x * blockDim.x + threadIdx.x;
    if (i < n) x[i] = fmaxf(x[i], 0.0f);
}

// ---------------------------------------------------------------------------
// Host-side orchestration
// ---------------------------------------------------------------------------
extern "C" void kernel_launch(void* const* d_in, const int* in_sizes, int n_in,
                              void* d_out, int out_size, void* d_ws, size_t ws_size,
                              hipStream_t stream) {
    const float* x    = (const float*)d_in[0];   // [N, 128]
    const int*   eidx = (const int*)  d_in[1];   // [2, E] flat
    const float* W1   = (const float*)d_in[2];   // [128, 128]
    const float* b1   = (const float*)d_in[3];   // [128]
    const float* W2   = (const float*)d_in[4];   // [128, 64]
    const float* b2   = (const float*)d_in[5];   // [64]
    float*       out  = (float*)d_out;           // [N, 64]

    const int N = NNODES;
    const int E = in_sizes[1] / 2;
    const int* src = eidx;
    const int* dst = eidx + E;

    // Workspace layout (bytes): dinv @ 0, h1 @ 1MB, out1 @ 53MB; h2 reuses h1.
    char*  ws   = (char*)d_ws;
    float* dinv = (float*)(ws);                              // N floats (deg -> dinv in place)
    float* h1   = (float*)(ws + (size_t)(1  << 20));         // N*128 floats (51.2 MB)
    float* out1 = (float*)(ws + (size_t)(53 << 20));         // N*128 floats (51.2 MB)
    float* h2   = h1;                                        // N*64 floats, reuse

    const int T = 256;
    constexpr int MB = 5;    // 100000 / (16*5) = 1250 M-blocks, exact

    // --- normalization ---
    init_deg_kernel<<<(N + T - 1) / T, T, 0, stream>>>(dinv, N);
    accum_deg_kernel<<<(E + T - 1) / T, T, 0, stream>>>(dst, dinv, E);
    deg_to_dinv_kernel<<<(N + T - 1) / T, T, 0, stream>>>(dinv, N);

    // --- layer 1: h1 = x @ W1 ; out1 = aggregate + b1 ; relu ---
    {
        const int jobs = (N / (16 * MB)) * (128 / 16);  // 1250*8 = 10000, /4 waves exact
        gemm_f32_wmma_kernel<MB><<<jobs / 4, 128, 0, stream>>>(x, W1, h1, N, 128, 128);
    }
    agg_init_kernel<128><<<((size_t)N * 128 + T - 1) / T, T, 0, stream>>>(h1, dinv, b1, out1, N);
    agg_edges_kernel<128><<<((size_t)E * 32 + T - 1) / T, T, 0, stream>>>(src, dst, dinv, h1, out1, E);
    relu_kernel<<<((size_t)N * 128 + T - 1) / T, T, 0, stream>>>(out1, N * 128);

    // --- layer 2: h2 = out1 @ W2 ; out = aggregate + b2 ---
    {
        const int jobs = (N / (16 * MB)) * (64 / 16);   // 1250*4 = 5000, /4 waves exact
        gemm_f32_wmma_kernel<MB><<<jobs / 4, 128, 0, stream>>>(out1, W2, h2, N, 64, 128);
    }
    agg_init_kernel<64><<<((size_t)N * 64 + T - 1) / T, T, 0, stream>>>(h2, dinv, b2, out, N);
    agg_edges_kernel<64><<<((size_t)E * 32 + T - 1) / T, T, 0, stream>>>(src, dst, dinv, h2, out, E);
}